// CASSI_27281632264566
// MI455X (gfx1250) — compile-verified
//
#include <hip/hip_runtime.h>
#include <stdint.h>

// ---------------------------------------------------------------------------
// CASSI A^T(A(x)) for MI455X (gfx1250). HBM-bound (AI ~0.5 flop/B, ~480 MB
// total traffic -> ~20us floor @ 23.3 TB/s).
//  - TDM tensor_load_to_lds streams x[b,:,m,:] into a PADDED LDS layout using
//    the D# pad feature (row stride 384 f32; the 128-DWORD inter-row pad is
//    the zero guard band for the +/-2l dispersion shifts).
//  - 28-band shift-reduction done with chained V_WMMA_F32_16X16X4_F32
//    (exact fp32; A carries shifted y1 values, B = ones => D rows = band sums).
//  - Unconditional ds_loads (guard bands hold zeros) -> no EXEC branches,
//    one dscnt wait per tile. B128 coalesced stores for the output.
// ---------------------------------------------------------------------------

typedef __attribute__((ext_vector_type(2))) float v2f;
typedef __attribute__((ext_vector_type(8))) float v8f;
typedef __attribute__((ext_vector_type(4))) unsigned int v4u;
typedef __attribute__((ext_vector_type(8))) int v8i;
typedef __attribute__((ext_vector_type(4))) int v4i;

#define LBANDS 28
#define MROWS 256
#define NCOLS 256
#define NTILES 20          // 20*16 = 320 >= NOUT=310
#define Y2PAD 320
#define ROWSTRIDE 384      // 256 data + 128 pad DWORDs (TDM pad feature)
#define GUARD 64           // front guard before row 0 (covers n >= -54)

// One TDM 2-D tile load: rows x row_elems f32, global row stride
// row_stride_elems, into LDS at lds_byte_off. pad_bits ORs the D# group1
// word0 pad controls (pad_enable/interval/amount). 6-arg builtin form.
__device__ __forceinline__ void tdm_load_2d(uint32_t lds_byte_off,
                                            const float* gptr,
                                            uint32_t row_elems,
                                            uint32_t rows,
                                            uint32_t row_stride_elems,
                                            uint32_t pad_bits) {
  uint64_t ga = (uint64_t)(uintptr_t)gptr;
  v4u g0;
  g0[0] = 1u;                                   // count=1, user descriptor
  g0[1] = lds_byte_off;                         // lds_addr [63:32]
  g0[2] = (uint32_t)(ga & 0xFFFFFFFFu);         // global_addr [95:64]
  g0[3] = (uint32_t)((ga >> 32) & 0x01FFFFFFu)  // global_addr [120:96]
          | (2u << 30);                         // type=2 ("image")
  v8i g1;
  g1[0] = (int)((2u << 16) | pad_bits);                 // data_size=4B + pad ctl
  g1[1] = (int)((row_elems & 0xFFFFu) << 16);           // tensor_dim0[15:0]
  g1[2] = (int)((row_elems >> 16) & 0xFFFFu)            // tensor_dim0[31:16]
          | (int)((rows & 0xFFFFu) << 16);              // tensor_dim1[15:0]
  g1[3] = (int)((rows >> 16) & 0xFFFFu)                 // tensor_dim1[31:16]
          | (int)((row_elems & 0xFFFFu) << 16);         // tile_dim0
  g1[4] = (int)(rows & 0xFFFFu);                        // tile_dim1 (tile_dim2=0)
  g1[5] = (int)row_stride_elems;                        // tensor_dim0_stride[31:0]
  g1[6] = 0;
  g1[7] = 0;
  v4i z4 = {0, 0, 0, 0};
  v8i z8 = {0, 0, 0, 0, 0, 0, 0, 0};
  __builtin_amdgcn_tensor_load_to_lds(g0, g1, z4, z4, z8, 0);
}

__global__ __launch_bounds__(256) void cassi_ata_kernel(
    const float* __restrict__ x,    // [B, L, M, N]
    const float* __restrict__ phi,  // [L, M, N]
    float* __restrict__ out) {      // [B, L, M, N]
  // Row l element n lives at y1buf[GUARD + l*ROWSTRIDE + n]; n in [-GUARD- ,320)
  // reads either the front guard or the previous row's zeroed pad.
  __shared__ __align__(16) float y1buf[GUARD + LBANDS * ROWSTRIDE];  // 43.3 KB
  __shared__ __align__(16) float plds[LBANDS * NCOLS];               // 28 KB
  __shared__ __align__(16) float y2[Y2PAD];                          // 1.25 KB

  const int m = blockIdx.x;
  const int b = blockIdx.y;
  const int tid = threadIdx.x;    // 256 threads = 8 wave32s

  // ---- Phase 0a: issue TDM loads (wave 0) ----
  if (tid == 0) {
    const float* xbase = x + (size_t)b * LBANDS * MROWS * NCOLS + (size_t)m * NCOLS;
    const float* pbase = phi + (size_t)m * NCOLS;
    // pad_enable(b20)=1, pad_interval(b24:22)=7 (256 DW), pad_amount(b31:25)=127 (128 DW)
    const uint32_t padctl = (1u << 20) | (7u << 22) | (127u << 25);
    tdm_load_2d((uint32_t)(uintptr_t)&y1buf[GUARD], xbase, NCOLS, LBANDS,
                MROWS * NCOLS, padctl);
    tdm_load_2d((uint32_t)(uintptr_t)&plds[0], pbase, NCOLS, LBANDS,
                MROWS * NCOLS, 0u);
  }

  // ---- Phase 0b: zero the guard/pad regions (disjoint from TDM targets) ----
  // front guard [0,GUARD) + per-row pads [GUARD + l*384 + 256, +384): 3648 f32.
  for (int idx = tid; idx < GUARD + LBANDS * 128; idx += 256) {
    int pos;
    if (idx < GUARD) {
      pos = idx;
    } else {
      int k = idx - GUARD;
      pos = GUARD + (k >> 7) * ROWSTRIDE + 256 + (k & 127);
    }
    y1buf[pos] = 0.0f;
  }

  if (tid == 0) __builtin_amdgcn_s_wait_tensorcnt(0);
  __syncthreads();

  // ---- Phase 0c: in-place mask: y1 = x * phi ----
#pragma unroll
  for (int l = 0; l < LBANDS; ++l) {
    y1buf[GUARD + l * ROWSTRIDE + tid] *= plds[l * NCOLS + tid];
  }
  __syncthreads();

  // ---- Phase 1: y2[c] = sum_l y1[l, c-2l] via WMMA f32 16x16x4 ----
  // A layout (16x4 f32): lanes 0-15 hold K=0,1; lanes 16-31 hold K=2,3;
  // A's M index (lane%16) = y2 column within the 16-wide tile. B = ones.
  const int wave = tid >> 5;
  const int lane = tid & 31;
  const int mcol = lane & 15;
  const int khalf = lane >> 4;
  const v2f ones = {1.0f, 1.0f};

  for (int t = wave; t < NTILES; t += 8) {
    const int c = t * 16 + mcol;
    v8f acc = {0.f, 0.f, 0.f, 0.f, 0.f, 0.f, 0.f, 0.f};
#pragma unroll
    for (int q = 0; q < 7; ++q) {
      const int l0 = 4 * q + 2 * khalf;          // bands l0, l0+1 this lane
      const int i0 = GUARD + l0 * ROWSTRIDE + (c - 2 * l0);
      v2f a = {y1buf[i0], y1buf[i0 + ROWSTRIDE - 2]};  // unconditional: guards=0
      acc = __builtin_amdgcn_wmma_f32_16x16x4_f32(
          false, a, false, ones, (short)0, acc, false, false);
    }
    // C/D layout: lane<16 VGPR r = D[r,lane] = y2[t*16+r]; lane>=16 -> +8.
    if (lane == 0) {
#pragma unroll
      for (int r = 0; r < 8; ++r) y2[t * 16 + r] = acc[r];
    } else if (lane == 16) {
#pragma unroll
      for (int r = 0; r < 8; ++r) y2[t * 16 + 8 + r] = acc[r];
    }
  }
  __syncthreads();

  // ---- Phase 2: out[l,n] = phi[l,m,n] * y2[n+2l], B128 coalesced stores ----
  const int lsel = tid >> 6;        // 0..3
  const int n = (tid & 63) * 4;     // 0,4,...,252
#pragma unroll
  for (int w = 0; w < 7; ++w) {
    const int l = w * 4 + lsel;
    const int cb = n + 2 * l;       // even -> 8B aligned y2 reads
    const float2 ya = *(const float2*)&y2[cb];
    const float2 yb = *(const float2*)&y2[cb + 2];
    const float4 p = *(const float4*)&plds[l * NCOLS + n];
    float4 o;
    o.x = ya.x * p.x;
    o.y = ya.y * p.y;
    o.z = yb.x * p.z;
    o.w = yb.y * p.w;
    *(float4*)&out[(size_t)((b * LBANDS + l) * MROWS + m) * NCOLS + n] = o;
  }
}

extern "C" void kernel_launch(void* const* d_in, const int* in_sizes, int n_in,
                              void* d_out, int out_size, void* d_ws, size_t ws_size,
                              hipStream_t stream) {
  const float* x = (const float*)d_in[0];
  const float* phi = (const float*)d_in[1];
  float* out = (float*)d_out;
  const int B = in_sizes[0] / (LBANDS * MROWS * NCOLS);  // 32
  dim3 grid(MROWS, B);
  cassi_ata_kernel<<<grid, 256, 0, stream>>>(x, phi, out);
}